// Model_83949430767933
// MI455X (gfx1250) — compile-verified
//
#include <hip/hip_runtime.h>
#include <math.h>

typedef float v2f __attribute__((ext_vector_type(2)));
typedef float v8f __attribute__((ext_vector_type(8)));

#define EMBD 128
#define OUTW 512   // EMB * (1 + 3 layers)
#define MROWS 32   // node-rows per gemm block (2 M-tiles per wave)
#define KQS   132  // LDS dwords per K-quad row: 32 rows * 4 + 4 pad (bank-perfect)

// ---------------------------------------------------------------------------
// SpMM scatter: one wave (32 lanes) per edge. out[r,:] += val * x[c,:]
// e is wave-uniform: force it scalar so rows/cols/vals use the SMEM path,
// leaving the vector pipe for the 512B row gather + 4 atomic f32 adds/lane.
// ---------------------------------------------------------------------------
__global__ __launch_bounds__(256) void spmm_scatter(
    const int* __restrict__ rows, const int* __restrict__ cols,
    const float* __restrict__ vals, const float* __restrict__ x,
    float* __restrict__ out, int nEdges)
{
    int gid = blockIdx.x * 256 + threadIdx.x;
    int e   = gid >> 5;                       // same for all 32 lanes of a wave
    if (e >= nEdges) return;                  // wave-uniform branch
    e = __builtin_amdgcn_readfirstlane(e);    // -> scalar loads below
    int lane = threadIdx.x & 31;
    int r   = rows[e];
    int c   = cols[e];
    float v = vals[e];
    float4 xv = *reinterpret_cast<const float4*>(x + (size_t)c * EMBD + lane * 4);
    float* o = out + (size_t)r * EMBD + lane * 4;
    atomicAdd(o + 0, v * xv.x);
    atomicAdd(o + 1, v * xv.y);
    atomicAdd(o + 2, v * xv.z);
    atomicAdd(o + 3, v * xv.w);
}

// ---------------------------------------------------------------------------
// X2 = l2norm(aggX + X); pre <- X2 ; d_out[:,0:128] <- 2*X2
// One wave per row, shfl_xor reduction (wave32), v_rcp for the reciprocal.
// ---------------------------------------------------------------------------
__global__ __launch_bounds__(256) void x2_norm(
    const float* __restrict__ agg, const float* __restrict__ X,
    float* __restrict__ pre, float* __restrict__ outp, int N)
{
    int gid  = blockIdx.x * 256 + threadIdx.x;
    int row  = gid >> 5;
    if (row >= N) return;
    int lane = threadIdx.x & 31;
    float4 a = *reinterpret_cast<const float4*>(agg + (size_t)row * EMBD + lane * 4);
    float4 x = *reinterpret_cast<const float4*>(X   + (size_t)row * EMBD + lane * 4);
    float4 v = make_float4(a.x + x.x, a.y + x.y, a.z + x.z, a.w + x.w);
    float ss = v.x * v.x + v.y * v.y + v.z * v.z + v.w * v.w;
    #pragma unroll
    for (int m = 16; m > 0; m >>= 1) ss += __shfl_xor(ss, m, 32);
    float inv = __builtin_amdgcn_rcpf(fmaxf(sqrtf(ss), 1e-12f));
    float4 n = make_float4(v.x * inv, v.y * inv, v.z * inv, v.w * inv);
    *reinterpret_cast<float4*>(pre + (size_t)row * EMBD + lane * 4) = n;
    float4 t = make_float4(2.f * n.x, 2.f * n.y, 2.f * n.z, 2.f * n.w);
    *reinterpret_cast<float4*>(outp + (size_t)row * OUTW + lane * 4) = t;
}

// ---------------------------------------------------------------------------
// Fused layer: emb = sigmoid((pre+aggA)@W2 + b2) + leaky((pre*aggS)@W3 + b3)
// Block: 32 node-rows, 8 waves. Wave w owns output cols [w*16, w*16+16) and
// computes TWO 16x16 M-tiles, reusing the same W2/W3 B-operand registers
// (4 global B loads -> 4 v_wmma_f32_16x16x4_f32 per K-step).
//
// LDS layout is K-quad interleaved: Ut[(k>>2)*KQS + m*4 + (k&3)], so the
// per-lane A fragment {K=k0+2*hi, K=k0+2*hi+1} for row m is ONE aligned
// ds_load_b64 into an even VGPR pair (no repack movs), and both the staging
// writes and the WMMA reads are LDS-bank conflict-free.
// Emb is written in-place over aggA (rows read before the barrier).
// ---------------------------------------------------------------------------
__global__ __launch_bounds__(256) void gemm_act(
    const float* __restrict__ pre, const float* __restrict__ aggA,
    const float* __restrict__ aggS,
    const float* __restrict__ W2, const float* __restrict__ b2,
    const float* __restrict__ W3, const float* __restrict__ b3,
    float* __restrict__ embOut, int N)
{
    __shared__ float Ut[(EMBD / 4) * KQS];   // 32 K-quads * 132 dwords
    __shared__ float Vt[(EMBD / 4) * KQS];

    const int tid     = threadIdx.x;
    const int rowBase = blockIdx.x * MROWS;

    #pragma unroll
    for (int i = 0; i < 16; ++i) {
        int elem = tid + i * 256;       // 32 rows x 128 cols = 4096 elems
        int m    = elem >> 7;
        int k    = elem & 127;
        int row  = rowBase + m;
        int rc   = row < N ? row : N - 1;
        int idx  = (k >> 2) * KQS + m * 4 + (k & 3);
        float p  = pre [(size_t)rc * EMBD + k];
        Ut[idx] = p + aggA[(size_t)rc * EMBD + k];
        Vt[idx] = p * aggS[(size_t)rc * EMBD + k];
    }
    __syncthreads();

    const int wave = tid >> 5;
    const int lane = tid & 31;
    const int nb   = wave * 16;        // output column base for this wave
    const int nn   = lane & 15;        // N index within tile
    const int hi   = lane >> 4;        // lane group: selects K+2 / M+8 halves

    v8f acc2[2], acc3[2];
    acc2[0] = acc2[1] = (v8f){0.f,0.f,0.f,0.f,0.f,0.f,0.f,0.f};
    acc3[0] = acc3[1] = (v8f){0.f,0.f,0.f,0.f,0.f,0.f,0.f,0.f};

    #pragma unroll 4
    for (int k0 = 0; k0 < EMBD; k0 += 4) {
        int ka = k0 + 2 * hi;          // A/B layout: vgpr v -> K = k0 + 2*hi + v
        v2f bA, bB;
        bA[0] = W2[(ka    ) * EMBD + nb + nn];
        bA[1] = W2[(ka + 1) * EMBD + nb + nn];
        bB[0] = W3[(ka    ) * EMBD + nb + nn];
        bB[1] = W3[(ka + 1) * EMBD + nb + nn];
        const int qbase = (k0 >> 2) * KQS + 2 * hi;
        #pragma unroll
        for (int t = 0; t < 2; ++t) {  // two M-tiles share bA/bB
            v2f aU = *reinterpret_cast<const v2f*>(&Ut[qbase + (t * 16 + nn) * 4]);
            v2f aV = *reinterpret_cast<const v2f*>(&Vt[qbase + (t * 16 + nn) * 4]);
            acc2[t] = __builtin_amdgcn_wmma_f32_16x16x4_f32(
                          false, aU, false, bA, (short)0, acc2[t], false, false);
            acc3[t] = __builtin_amdgcn_wmma_f32_16x16x4_f32(
                          false, aV, false, bB, (short)0, acc3[t], false, false);
        }
    }

    const float bias2 = b2[nb + nn];
    const float bias3 = b3[nb + nn];
    #pragma unroll
    for (int t = 0; t < 2; ++t) {
        #pragma unroll
        for (int j = 0; j < 8; ++j) {
            int m   = t * 16 + j + 8 * hi;   // C/D layout: vgpr j -> M = j + 8*hi
            int row = rowBase + m;
            if (row >= N) continue;
            float h2 = acc2[t][j] + bias2;
            float h3 = acc3[t][j] + bias3;
            float s  = __builtin_amdgcn_rcpf(1.0f + __expf(-h2));  // sigmoid
            float l  = h3 > 0.0f ? h3 : 0.01f * h3;                // leaky relu
            embOut[(size_t)row * EMBD + nb + nn] = s + l;
        }
    }
}

// ---------------------------------------------------------------------------
// d_out[:, colOfs:colOfs+128] = l2norm(emb) ; one wave per row
// ---------------------------------------------------------------------------
__global__ __launch_bounds__(256) void row_norm_out(
    const float* __restrict__ emb, float* __restrict__ outp, int colOfs, int N)
{
    int gid  = blockIdx.x * 256 + threadIdx.x;
    int row  = gid >> 5;
    if (row >= N) return;
    int lane = threadIdx.x & 31;
    float4 v = *reinterpret_cast<const float4*>(emb + (size_t)row * EMBD + lane * 4);
    float ss = v.x * v.x + v.y * v.y + v.z * v.z + v.w * v.w;
    #pragma unroll
    for (int m = 16; m > 0; m >>= 1) ss += __shfl_xor(ss, m, 32);
    float inv = __builtin_amdgcn_rcpf(fmaxf(sqrtf(ss), 1e-12f));
    float4 n = make_float4(v.x * inv, v.y * inv, v.z * inv, v.w * inv);
    *reinterpret_cast<float4*>(outp + (size_t)row * OUTW + colOfs + lane * 4) = n;
}

// ---------------------------------------------------------------------------
extern "C" void kernel_launch(void* const* d_in, const int* in_sizes, int n_in,
                              void* d_out, int out_size, void* d_ws, size_t ws_size,
                              hipStream_t stream)
{
    const int*   Aidx = (const int*)  d_in[0];
    const float* Aval = (const float*)d_in[1];
    const int*   Sidx = (const int*)  d_in[2];
    const float* Sval = (const float*)d_in[3];
    const float* X    = (const float*)d_in[4];
    const float* W2   = (const float*)d_in[5];
    const float* b2   = (const float*)d_in[6];
    const float* W3   = (const float*)d_in[7];
    const float* b3   = (const float*)d_in[8];

    const int E  = in_sizes[0] / 2;
    const int ES = in_sizes[2] / 2;
    const int N  = in_sizes[4] / EMBD;

    const int* Arows = Aidx;  const int* Acols = Aidx + E;
    const int* Srows = Sidx;  const int* Scols = Sidx + ES;

    const size_t mat = (size_t)N * EMBD;
    float* B0 = (float*)d_ws;        // pre
    float* B1 = B0 + mat;            // allAgg / emb
    float* B2 = B1 + mat;            // subAgg
    float* out = (float*)d_out;

    const int spmmBlkA = (int)(((long long)E  * 32 + 255) / 256);
    const int spmmBlkS = (int)(((long long)ES * 32 + 255) / 256);
    const int rowBlk   = (int)(((long long)N  * 32 + 255) / 256);
    const int gemmBlk  = (N + MROWS - 1) / MROWS;

    // ---- X2 = l2norm(A@X + X) ; out[:,0:128] = 2*X2 -------------------------
    hipMemsetAsync(B1, 0, mat * sizeof(float), stream);
    spmm_scatter<<<spmmBlkA, 256, 0, stream>>>(Arows, Acols, Aval, X, B1, E);
    x2_norm<<<rowBlk, 256, 0, stream>>>(B1, X, B0, out, N);

    float* pre  = B0;
    float* aggA = B1;
    float* aggS = B2;

    for (int k = 0; k < 3; ++k) {
        hipMemsetAsync(aggA, 0, mat * sizeof(float), stream);
        hipMemsetAsync(aggS, 0, mat * sizeof(float), stream);
        spmm_scatter<<<spmmBlkA, 256, 0, stream>>>(Arows, Acols, Aval, pre, aggA, E);
        spmm_scatter<<<spmmBlkS, 256, 0, stream>>>(Srows, Scols, Sval, pre, aggS, ES);
        gemm_act<<<gemmBlk, 256, 0, stream>>>(
            pre, aggA, aggS,
            W2 + (size_t)k * EMBD * EMBD, b2 + (size_t)k * EMBD,
            W3 + (size_t)k * EMBD * EMBD, b3 + (size_t)k * EMBD,
            aggA /* emb written in-place over allAgg */, N);
        row_norm_out<<<rowBlk, 256, 0, stream>>>(aggA, out, (k + 1) * EMBD, N);
        float* t = pre; pre = aggA; aggA = t;   // emb becomes next pre
    }
}